// Convert2Dto3DWithPadding_52656299049241
// MI455X (gfx1250) — compile-verified
//
#include <hip/hip_runtime.h>
#include <stdint.h>

// Convert2Dto3DWithPadding for MI455X (gfx1250, wave32).
// Pure data movement (~376 MB, 0 FLOPs) -> HBM-bound (~16 us at 23.3 TB/s).
// Bulk copy uses the CDNA5 async global<->LDS DMA path (ASYNCcnt-tracked):
//   global_load_async_to_lds_b128   (memory -> LDS, no VGPR round trip)
//   global_store_async_from_lds_b128 (LDS -> memory)
// MLP: each wave batches R=8 rows (8 x 512 B async loads in flight) before a
// single s_wait_asynccnt 0, then fires 8 async stores that drain at s_endpgm.

#define WAVES_PER_BLOCK   8
#define THREADS_PER_BLOCK 256
#define ROWS_PER_WAVE     8          // R: 4 KB in flight per wave
#define F_FAST            128        // fast-path feature count (512 B rows)

__device__ __forceinline__ void async_load_b128(unsigned lds_addr, const float* gsrc) {
  // VDST = LDS byte address (VGPR), VADDR = 64-bit global address (VGPR pair)
  asm volatile("global_load_async_to_lds_b128 %0, %1, off"
               :
               : "v"(lds_addr), "v"((unsigned long long)(uintptr_t)gsrc)
               : "memory");
}

__device__ __forceinline__ void async_store_b128(float* gdst, unsigned lds_addr) {
  // VADDR = 64-bit global address (VGPR pair), VSRC = LDS byte address (VGPR)
  asm volatile("global_store_async_from_lds_b128 %0, %1, off"
               :
               : "v"((unsigned long long)(uintptr_t)gdst), "v"(lds_addr)
               : "memory");
}

__device__ __forceinline__ void wait_async_0() {
#if __has_builtin(__builtin_amdgcn_s_wait_asynccnt)
  __builtin_amdgcn_s_wait_asynccnt(0);
#else
  asm volatile("s_wait_asynccnt 0x0" ::: "memory");
#endif
}

__device__ __forceinline__ int lower_bound_i32(const int* __restrict__ a, int n, int key) {
  int lo = 0, hi = n;
  while (lo < hi) {
    int mid = (lo + hi) >> 1;
    if (a[mid] < key) lo = mid + 1; else hi = mid;
  }
  return lo;
}

// ---------------------------------------------------------------------------
// Fast path: F == 128 (one b128 chunk per lane per row).
// Each wave handles ROWS_PER_WAVE consecutive output rows.
// ---------------------------------------------------------------------------
__global__ void __launch_bounds__(THREADS_PER_BLOCK)
pad_gather_f128(const float* __restrict__ x,
                const int* __restrict__ batch_ids,
                const int* __restrict__ g_max_atoms,
                float* __restrict__ out,
                int N, int rows) {
  __shared__ float lds_buf[WAVES_PER_BLOCK * ROWS_PER_WAVE * F_FAST];  // 32 KB

  const int tid  = (int)threadIdx.x;
  const int wave = tid >> 5;           // wave32
  const int lane = tid & 31;

  const int w  = (int)blockIdx.x * WAVES_PER_BLOCK + wave;
  const int r0 = w * ROWS_PER_WAVE;
  if (r0 >= rows) return;

  const int M = g_max_atoms[0];        // device-side scalar (=max_num_atoms)

  // One division per wave; (g, m) advanced incrementally over 8 rows.
  int g = r0 / M;
  int m = r0 - g * M;

  // Per-graph extent from sorted batch_ids (wave-uniform, L2-resident).
  // Crossing into graph g+1 reuses: start_new = end_old (one search/crossing).
  int start = lower_bound_i32(batch_ids, N, g);
  int end   = lower_bound_i32(batch_ids, N, g + 1);

  const unsigned lds_w =
      (unsigned)(uintptr_t)&lds_buf[wave * ROWS_PER_WAVE * F_FAST] +
      (unsigned)(lane << 4);           // +16 B per lane

  bool validv[ROWS_PER_WAVE];

  // ---- Phase 1: issue all async loads (4 KB/wave in flight), write mask ----
#pragma unroll
  for (int i = 0; i < ROWS_PER_WAVE; ++i) {
    const int r = r0 + i;
    bool valid = false;
    if (r < rows) {
      valid = (m < (end - start));
      if (lane == 0)
        out[(size_t)rows * F_FAST + (size_t)r] = valid ? 1.0f : 0.0f;
      if (valid) {
        const float* src =
            x + (((size_t)(start + m)) << 7) + (size_t)(lane << 2);
        async_load_b128(lds_w + (unsigned)(i * (F_FAST * 4)), src);
      }
    }
    validv[i] = valid;
    if (i + 1 < ROWS_PER_WAVE) {
      if (++m == M) {                  // graph boundary crossing
        m = 0;
        ++g;
        start = end;
        end = lower_bound_i32(batch_ids, N, g + 1);
      }
    }
  }

  wait_async_0();                      // all 8 row-chunks now resident in LDS

  // ---- Phase 2: fire async stores / zero padding; s_endpgm drains them ----
#pragma unroll
  for (int i = 0; i < ROWS_PER_WAVE; ++i) {
    const int r = r0 + i;
    if (r >= rows) break;
    float* dst = out + (((size_t)r) << 7) + (size_t)(lane << 2);
    if (validv[i]) {
      async_store_b128(dst, lds_w + (unsigned)(i * (F_FAST * 4)));
    } else {
      const float4 z = make_float4(0.0f, 0.0f, 0.0f, 0.0f);
      *reinterpret_cast<float4*>(dst) = z;   // global_store_b128
    }
  }
}

// ---------------------------------------------------------------------------
// Generic fallback for arbitrary F (plain VGPR copy, one wave per row).
// ---------------------------------------------------------------------------
__global__ void __launch_bounds__(THREADS_PER_BLOCK)
pad_gather_generic(const float* __restrict__ x,
                   const int* __restrict__ batch_ids,
                   const int* __restrict__ g_max_atoms,
                   float* __restrict__ out,
                   int N, int F, int rows) {
  const int tid  = (int)threadIdx.x;
  const int wave = tid >> 5;
  const int lane = tid & 31;

  const int r = (int)blockIdx.x * WAVES_PER_BLOCK + wave;
  if (r >= rows) return;

  const int M = g_max_atoms[0];
  const int g = r / M;
  const int m = r - g * M;

  const int start = lower_bound_i32(batch_ids, N, g);
  const int end   = lower_bound_i32(batch_ids, N, g + 1);
  const bool valid = (m < (end - start));

  if (lane == 0)
    out[(size_t)rows * (size_t)F + (size_t)r] = valid ? 1.0f : 0.0f;

  float* dst = out + (size_t)r * (size_t)F;
  const int Fv = F & ~3;
  if (valid) {
    const float* src = x + (size_t)(start + m) * (size_t)F;
    for (int c = (lane << 2); c < Fv; c += 4 * 32) {
      const float4 v = *reinterpret_cast<const float4*>(src + c);
      *reinterpret_cast<float4*>(dst + c) = v;
    }
    if (lane < (F - Fv)) dst[Fv + lane] = src[Fv + lane];
  } else {
    const float4 z = make_float4(0.0f, 0.0f, 0.0f, 0.0f);
    for (int c = (lane << 2); c < Fv; c += 4 * 32)
      *reinterpret_cast<float4*>(dst + c) = z;
    if (lane < (F - Fv)) dst[Fv + lane] = 0.0f;
  }
}

extern "C" void kernel_launch(void* const* d_in, const int* in_sizes, int n_in,
                              void* d_out, int out_size, void* d_ws, size_t ws_size,
                              hipStream_t stream) {
  (void)n_in; (void)d_ws; (void)ws_size;
  const float* x      = (const float*)d_in[0];   // [N, F] float32
  const int*   batch  = (const int*)d_in[1];     // [N] sorted graph ids
  // d_in[2] = num_graphs (device scalar, unused: g < G implied by r < rows)
  const int*   g_maxM = (const int*)d_in[3];     // max_num_atoms (device scalar)
  float*       out    = (float*)d_out;

  const int N    = in_sizes[1];
  const int F    = in_sizes[0] / in_sizes[1];    // 128 for the reference shapes
  const int rows = out_size / (F + 1);           // G*M (result G*M*F + mask G*M)

  if (F == F_FAST) {
    const int rows_per_block = WAVES_PER_BLOCK * ROWS_PER_WAVE;   // 64
    const int blocks = (rows + rows_per_block - 1) / rows_per_block;
    pad_gather_f128<<<blocks, THREADS_PER_BLOCK, 0, stream>>>(
        x, batch, g_maxM, out, N, rows);
  } else {
    const int blocks = (rows + WAVES_PER_BLOCK - 1) / WAVES_PER_BLOCK;
    pad_gather_generic<<<blocks, THREADS_PER_BLOCK, 0, stream>>>(
        x, batch, g_maxM, out, N, F, rows);
  }
}